// Normalized_Gradient_Field_46308337385672
// MI455X (gfx1250) — compile-verified
//
#include <hip/hip_runtime.h>
#include <stdint.h>

// Volume dims (1,1,192,224,192) fp32
#define D_ 192
#define H_ 224
#define W_ 192
// Output tile per block
#define TD 8
#define TH 16
#define TW 32
// Input (haloed) tile
#define LD (TD + 2)          // 10
#define LH (TH + 2)          // 18
#define LW (TW + 2)          // 34
#define PLANE (LH * LW)      // 612
#define TILE_N (LD * PLANE)  // 6120 floats per volume
#define NTHREADS 256
#define EPS2 0.01f

typedef float v2f __attribute__((ext_vector_type(2)));
typedef float v8f __attribute__((ext_vector_type(8)));

// Low 32 bits of a generic pointer into LDS == DS byte offset on gfx12
// (aperture tag lives entirely in the high 32 bits; HW adds LDS_BASE).
__device__ __forceinline__ unsigned lds_off_u32(const void* p) {
  return (unsigned)(uintptr_t)p;
}

// Async global -> LDS copy, 4B per lane, tracked with ASYNCcnt (gfx1250).
__device__ __forceinline__ void async_ld_b32(unsigned lds_byte, const float* g) {
  asm volatile("global_load_async_to_lds_b32 %0, %1, off"
               :
               : "v"(lds_byte), "v"((unsigned long long)(uintptr_t)g)
               : "memory");
}

__device__ __forceinline__ void wait_async_all() {
  asm volatile("s_wait_asynccnt 0x0" ::: "memory");
}

// Deterministic wave32 sum. Primary path: one V_WMMA_F32_16X16X4_F32.
// A 16x4 f32 layout (ISA 7.12.2): VGPR0 lane m (m<16) = A[m,0],
// VGPR0 lane m+16 = A[m,2]; VGPR1 = K1/K3 (set to 0). With B = ones,
// D[m,n] = acc_m + acc_{m+16}. Summing D's 8 VGPRs gives the half-wave
// sum per lane group; one shfl_xor(16) completes the full 32-lane sum.
__device__ __forceinline__ float wave_sum32(float x) {
#if __has_builtin(__builtin_amdgcn_wmma_f32_16x16x4_f32)
  v2f a;
  a[0] = x;
  a[1] = 0.0f;
  v2f b;
  b[0] = 1.0f;
  b[1] = 1.0f;
  v8f c = {};
  v8f d = __builtin_amdgcn_wmma_f32_16x16x4_f32(false, a, false, b, (short)0, c,
                                                false, false);
  float s = ((d[0] + d[1]) + (d[2] + d[3])) + ((d[4] + d[5]) + (d[6] + d[7]));
  s += __shfl_xor(s, 16, 32);
  return s;
#else
  for (int m = 16; m >= 1; m >>= 1) x += __shfl_xor(x, m, 32);
  return x;
#endif
}

// For one z-plane and one (h,w) column, read the 3x3 patch from LDS and
// produce: P = full 3x3 sum (for Iz), Rx = sum_h (right-left) (for Ix),
// Ry = bottom_row_sum - top_row_sum (for Iy).
__device__ __forceinline__ void plane_stats(const float* __restrict__ s, int i,
                                            float& P, float& Rx, float& Ry) {
  float x00 = s[i + 0], x01 = s[i + 1], x02 = s[i + 2];
  float x10 = s[i + LW + 0], x11 = s[i + LW + 1], x12 = s[i + LW + 2];
  float x20 = s[i + 2 * LW + 0], x21 = s[i + 2 * LW + 1], x22 = s[i + 2 * LW + 2];
  float r0 = x00 + x01 + x02;
  float r1 = x10 + x11 + x12;
  float r2 = x20 + x21 + x22;
  P = r0 + r1 + r2;
  Ry = r2 - r0;
  Rx = (x02 + x12 + x22) - (x00 + x10 + x20);
}

__global__ __launch_bounds__(NTHREADS) void ngf_stencil(
    const float* __restrict__ I, const float* __restrict__ J,
    float* __restrict__ blocksums, float* __restrict__ outp, int atomic_mode,
    float invN) {
  __shared__ float sI[TILE_N];
  __shared__ float sJ[TILE_N];
  __shared__ float wred[NTHREADS / 32];

  const int t = threadIdx.x;
  const int w0 = blockIdx.x * TW;
  const int h0 = blockIdx.y * TH;
  const int d0 = blockIdx.z * TD;

  const unsigned sI0 = lds_off_u32(&sI[0]);
  const unsigned sJ0 = lds_off_u32(&sJ[0]);

  // ---- Fill haloed tile via async global->LDS, clamped (edge replication) ----
  for (int e = t; e < TILE_N; e += NTHREADS) {
    int lw = e % LW;
    int r = e / LW;
    int lh = r % LH;
    int ld = r / LH;
    int gw = w0 - 1 + lw;
    gw = gw < 0 ? 0 : (gw > W_ - 1 ? W_ - 1 : gw);
    int gh = h0 - 1 + lh;
    gh = gh < 0 ? 0 : (gh > H_ - 1 ? H_ - 1 : gh);
    int gd = d0 - 1 + ld;
    gd = gd < 0 ? 0 : (gd > D_ - 1 ? D_ - 1 : gd);
    size_t gi = ((size_t)gd * H_ + gh) * (size_t)W_ + gw;
    unsigned lb = (unsigned)e * 4u;
    async_ld_b32(sI0 + lb, I + gi);
    async_ld_b32(sJ0 + lb, J + gi);
  }
  wait_async_all();
  __syncthreads();

  // ---- Stencil sweep: each thread owns (w = t&31, h in {t>>5, t>>5 + 8}),
  //      marching d = 0..7 with a 3-plane ring buffer. ----
  const int w = t & (TW - 1);
  const int hg = t >> 5;  // uniform within a wave32
  float acc = 0.0f;

#pragma unroll
  for (int hp = 0; hp < 2; ++hp) {
    const int hh = hg + hp * 8;
    const int cb = hh * LW + w;  // top-left of 3x3 patch inside a plane
    float PI[3], RXI[3], RYI[3], PJ[3], RXJ[3], RYJ[3];
    plane_stats(sI, 0 * PLANE + cb, PI[0], RXI[0], RYI[0]);
    plane_stats(sI, 1 * PLANE + cb, PI[1], RXI[1], RYI[1]);
    plane_stats(sJ, 0 * PLANE + cb, PJ[0], RXJ[0], RYJ[0]);
    plane_stats(sJ, 1 * PLANE + cb, PJ[1], RXJ[1], RYJ[1]);
#pragma unroll
    for (int d = 0; d < TD; ++d) {
      const int s0 = d % 3, s1 = (d + 1) % 3, s2 = (d + 2) % 3;
      plane_stats(sI, (d + 2) * PLANE + cb, PI[s2], RXI[s2], RYI[s2]);
      plane_stats(sJ, (d + 2) * PLANE + cb, PJ[s2], RXJ[s2], RYJ[s2]);
      float Ix = 0.5f * (RXI[s0] + RXI[s1] + RXI[s2]);
      float Iy = 0.5f * (RYI[s0] + RYI[s1] + RYI[s2]);
      float Iz = 0.5f * (PI[s2] - PI[s0]);
      float Jx = 0.5f * (RXJ[s0] + RXJ[s1] + RXJ[s2]);
      float Jy = 0.5f * (RYJ[s0] + RYJ[s1] + RYJ[s2]);
      float Jz = 0.5f * (PJ[s2] - PJ[s0]);
      float Im = Ix * Ix + Iy * Iy + Iz * Iz + EPS2;
      float Jm = Jx * Jx + Jy * Jy + Jz * Jz + EPS2;
      float dt = Ix * Jx + Iy * Jy + Iz * Jz;
      acc += 1.0f - (dt * dt) / (Im * Jm);
    }
  }

  // ---- Block reduction: WMMA wave sum, then 8 wave sums in fixed order ----
  float wsum = wave_sum32(acc);
  const int lane = t & 31, wid = t >> 5;
  if (lane == 0) wred[wid] = wsum;
  __syncthreads();
  if (t == 0) {
    float tot = wred[0];
#pragma unroll
    for (int i = 1; i < NTHREADS / 32; ++i) tot += wred[i];
    const int bid = (blockIdx.z * gridDim.y + blockIdx.y) * gridDim.x + blockIdx.x;
    if (atomic_mode) {
      atomicAdd(outp, tot * invN);
    } else {
      blocksums[bid] = tot;
    }
  }
}

__global__ void ngf_zero(float* __restrict__ o) {
  if (threadIdx.x == 0 && blockIdx.x == 0) o[0] = 0.0f;
}

__global__ __launch_bounds__(NTHREADS) void ngf_finalize(
    const float* __restrict__ bs, float* __restrict__ o, int n, float invN) {
  __shared__ double red[NTHREADS];
  double s = 0.0;
  for (int i = threadIdx.x; i < n; i += NTHREADS) s += (double)bs[i];
  red[threadIdx.x] = s;
  __syncthreads();
  for (int k = NTHREADS / 2; k > 0; k >>= 1) {
    if ((int)threadIdx.x < k) red[threadIdx.x] += red[threadIdx.x + k];
    __syncthreads();
  }
  if (threadIdx.x == 0) o[0] = (float)(red[0] * (double)invN);
}

extern "C" void kernel_launch(void* const* d_in, const int* in_sizes, int n_in,
                              void* d_out, int out_size, void* d_ws,
                              size_t ws_size, hipStream_t stream) {
  (void)in_sizes;
  (void)n_in;
  (void)out_size;
  const float* I = (const float*)d_in[0];
  const float* J = (const float*)d_in[1];
  float* out = (float*)d_out;
  float* ws = (float*)d_ws;

  dim3 grid(W_ / TW, H_ / TH, D_ / TD);  // 6 x 14 x 24 = 2016 blocks
  const int nblocks = (int)(grid.x * grid.y * grid.z);
  const float invN = 1.0f / (float)(D_ * H_ * W_);

  if (ws != nullptr && ws_size >= (size_t)nblocks * sizeof(float)) {
    // Deterministic two-pass reduction.
    ngf_stencil<<<grid, NTHREADS, 0, stream>>>(I, J, ws, out, 0, invN);
    ngf_finalize<<<1, NTHREADS, 0, stream>>>(ws, out, nblocks, invN);
  } else {
    // Fallback: atomic accumulation directly into d_out.
    ngf_zero<<<1, 32, 0, stream>>>(out);
    ngf_stencil<<<grid, NTHREADS, 0, stream>>>(I, J, ws, out, 1, invN);
  }
}